// MultiheadAttention_29850022707411
// MI455X (gfx1250) — compile-verified
//
#include <hip/hip_runtime.h>
#include <stdint.h>

// CDNA5 (gfx1250) wave32 WMMA GEMM with async global->LDS double buffering.
// Computes C[m,n] = sum_k A[m,k] * W[map(n),k] + bias[map(n)]
// (NT GEMM: A is MxK row-major, W row-major with K contiguous.)
//
// The reference MHA reduces algebraically to two such GEMMs because
// softmax(...).sum(-1) == 1, making the attention an identity on V:
//   V   = x @ Wv^T + bv     (Wv = V-rows of Wqkv: h*192+128 .. h*192+191)
//   out = V @ Wo^T + bo

typedef __attribute__((ext_vector_type(2))) float v2f;
typedef __attribute__((ext_vector_type(8))) float v8f;

#define TILE_M 128
#define TILE_N 128
#define TILE_K 32
#define LDS_STRIDE 36   // 32 + 4 pad: conflict-free b64 frag reads, keeps 16B alignment

// Issue one K-tile stage: 4 async b128 loads for A + 4 for W per thread.
// Per wave: 8 instructions x 32 lanes x 16B = 4KB; 8 waves cover the 32KB tile pair.
__device__ __forceinline__ void issue_stage(const float* __restrict__ A,
                                            const float* __restrict__ W,
                                            int K, int kt, int remap,
                                            int blockM, int blockN, int tid,
                                            float* AldsBuf, float* BldsBuf)
{
    #pragma unroll
    for (int j = 0; j < 4; ++j) {
        const int idx = tid + j * 256;     // 0..1023 ; 8 float4-chunks per 32-float row
        const int row = idx >> 3;
        const int ch  = (idx & 7) * 4;

        const uint64_t ga = (uint64_t)(uintptr_t)&A[(size_t)(blockM + row) * K + kt + ch];
        const uint32_t la = (uint32_t)(uintptr_t)&AldsBuf[row * LDS_STRIDE + ch];
        asm volatile("global_load_async_to_lds_b128 %0, %1, off"
                     :: "v"(la), "v"(ga) : "memory");

        const int nrow = blockN + row;
        const int wrow = remap ? ((nrow >> 6) * 192 + 128 + (nrow & 63)) : nrow;
        const uint64_t gb = (uint64_t)(uintptr_t)&W[(size_t)wrow * K + kt + ch];
        const uint32_t lb = (uint32_t)(uintptr_t)&BldsBuf[row * LDS_STRIDE + ch];
        asm volatile("global_load_async_to_lds_b128 %0, %1, off"
                     :: "v"(lb), "v"(gb) : "memory");
    }
}

__global__ __launch_bounds__(256)
void gemm_nt_bias_wmma_f32(const float* __restrict__ A,    // M x K
                           const float* __restrict__ W,    // rows x K (row selected by map)
                           const float* __restrict__ bias, // indexed by map(n)
                           float* __restrict__ C,          // M x N
                           int M, int N, int K, int remap)
{
    __shared__ float Alds[2][TILE_M * LDS_STRIDE];
    __shared__ float Blds[2][TILE_N * LDS_STRIDE];

    const int tid  = threadIdx.x;
    const int lane = tid & 31;
    const int wave = tid >> 5;
    const int l15  = lane & 15;
    const int hi   = lane >> 4;          // 0: K pair {0,1} / rows 0-7 ; 1: K pair {2,3} / rows 8-15

    const int waveN = wave & 3;          // 4 waves across N -> 32-col slab each
    const int waveM = wave >> 2;         // 2 waves across M -> 64-row slab each

    const int blockM = blockIdx.x * TILE_M;
    const int blockN = blockIdx.y * TILE_N;

    v8f acc[4][2];
    #pragma unroll
    for (int i = 0; i < 4; ++i)
        #pragma unroll
        for (int j = 0; j < 2; ++j)
            acc[i][j] = (v8f){0.f,0.f,0.f,0.f,0.f,0.f,0.f,0.f};

    // Prologue: async-load the first K-tile into buffer 0.
    issue_stage(A, W, K, 0, remap, blockM, blockN, tid, Alds[0], Blds[0]);

    int buf = 0;
    for (int kt = 0; kt < K; kt += TILE_K, buf ^= 1) {
        // Issue next tile into the other buffer, then wait for the CURRENT tile.
        // Async loads complete in order, so ASYNCcnt <= 8 (= next stage still in
        // flight) guarantees this stage's 8 loads have landed in LDS.
        if (kt + TILE_K < K) {
            issue_stage(A, W, K, kt + TILE_K, remap, blockM, blockN, tid,
                        Alds[buf ^ 1], Blds[buf ^ 1]);
            asm volatile("s_wait_asynccnt 0x8" ::: "memory");
        } else {
            asm volatile("s_wait_asynccnt 0x0" ::: "memory");
        }
        __syncthreads();   // publish this tile to all waves in the workgroup

        // ---- 8 K-steps of 4, v_wmma_f32_16x16x4_f32 ----
        #pragma unroll
        for (int ks = 0; ks < 8; ++ks) {
            const int kcol = ks * 4 + 2 * hi;  // per-lane K pair (contiguous float2)

            v2f aF[4], bF[2];
            #pragma unroll
            for (int i = 0; i < 4; ++i)
                aF[i] = *reinterpret_cast<const v2f*>(
                    &Alds[buf][(waveM * 64 + i * 16 + l15) * LDS_STRIDE + kcol]);
            #pragma unroll
            for (int j = 0; j < 2; ++j)
                bF[j] = *reinterpret_cast<const v2f*>(
                    &Blds[buf][(waveN * 32 + j * 16 + l15) * LDS_STRIDE + kcol]);

            #pragma unroll
            for (int i = 0; i < 4; ++i)
                #pragma unroll
                for (int j = 0; j < 2; ++j)
                    acc[i][j] = __builtin_amdgcn_wmma_f32_16x16x4_f32(
                        /*neg_a=*/false, aF[i],
                        /*neg_b=*/false, bF[j],
                        /*c_mod=*/(short)0, acc[i][j],
                        /*reuse_a=*/false, /*reuse_b=*/false);
        }

        __syncthreads();   // all waves done reading buf before it is re-issued
    }

    // ---- bias add + store (D layout: VGPR r -> row base + 8*hi + r, col = l15) ----
    #pragma unroll
    for (int j = 0; j < 2; ++j) {
        const int n    = blockN + waveN * 32 + j * 16 + l15;
        const int brow = remap ? ((n >> 6) * 192 + 128 + (n & 63)) : n;
        const float bv = bias[brow];
        #pragma unroll
        for (int i = 0; i < 4; ++i) {
            const int mbase = blockM + waveM * 64 + i * 16 + hi * 8;
            #pragma unroll
            for (int r = 0; r < 8; ++r)
                C[(size_t)(mbase + r) * N + n] = acc[i][j][r] + bv;
        }
    }
}

extern "C" void kernel_launch(void* const* d_in, const int* in_sizes, int n_in,
                              void* d_out, int out_size, void* d_ws, size_t ws_size,
                              hipStream_t stream) {
    // setup_inputs() order: x, Wqkv, bqkv, Wo, bo  (all fp32)
    const float* x    = (const float*)d_in[0];  // (B,S,E) = (2,2048,1024)
    const float* Wqkv = (const float*)d_in[1];  // (3E, E) = (3072,1024)
    const float* bqkv = (const float*)d_in[2];  // (3E,)
    const float* Wo   = (const float*)d_in[3];  // (E, E)
    const float* bo   = (const float*)d_in[4];  // (E,)
    float* out = (float*)d_out;                 // (B,S,E)

    const int M = 2 * 2048;   // B*S
    const int N = 1024;       // E
    const int K = 1024;       // E

    float* V = (float*)d_ws;  // intermediate V projection: M*N floats = 16.8 MB

    dim3 grid(M / TILE_M, N / TILE_N);  // 32 x 8 blocks
    dim3 block(256);                    // 8 wave32s

    // Phase 1: V = x @ Wv^T + bv  (Wv rows: h*192+128 .. h*192+191 per head h)
    gemm_nt_bias_wmma_f32<<<grid, block, 0, stream>>>(x, Wqkv, bqkv, V, M, N, K, /*remap=*/1);
    // Phase 2: out = V @ Wo^T + bo
    gemm_nt_bias_wmma_f32<<<grid, block, 0, stream>>>(V, Wo, bo, out, M, N, K, /*remap=*/0);
}